// Encoder_21680994910290
// MI455X (gfx1250) — compile-verified
//
#include <hip/hip_runtime.h>

// ---------------------------------------------------------------------------
// Two-layer GCN for MI455X (gfx1250).
//  - GEMMs use V_WMMA_F32_16X16X4_F32 (exact fp32 matrix path, wave32).
//  - Edge gather/scatter: wave-per-edge, float4 gathers + global_atomic_add_f32.
//    h (51.2MB) and agg (51.2MB) both fit in the 192MB L2 -> random edge
//    traffic is L2-resident; HBM only sees the streaming passes.
// ---------------------------------------------------------------------------

typedef float v2f __attribute__((ext_vector_type(2)));
typedef float v8f __attribute__((ext_vector_type(8)));

#define NFEAT 128   // F == C == 128 per reference

// ---- degree / norm precompute ---------------------------------------------
__global__ void k_init_deg(float* deg, int n) {
  int i = blockIdx.x * blockDim.x + threadIdx.x;
  if (i < n) deg[i] = 1.0f;   // self-loop contribution
}

__global__ void k_degree(const int* __restrict__ dst, float* deg, int E) {
  int e = blockIdx.x * blockDim.x + threadIdx.x;
  if (e < E) atomicAdd(&deg[dst[e]], 1.0f);
}

__global__ void k_dinv(float* deg, int n) {
  int i = blockIdx.x * blockDim.x + threadIdx.x;
  if (i < n) {
    float d = deg[i];
    deg[i] = (d > 0.0f) ? rsqrtf(d) : 0.0f;  // in-place: deg -> dinv
  }
}

__global__ void k_norm(const int* __restrict__ src, const int* __restrict__ dst,
                       const float* __restrict__ dinv, float* __restrict__ norm, int E) {
  int e = blockIdx.x * blockDim.x + threadIdx.x;
  if (e < E) norm[e] = dinv[src[e]] * dinv[dst[e]];
}

// ---- dense transform: H = X @ W  via WMMA f32 16x16x4 ----------------------
// Block = 256 threads = 8 waves. Block b covers rows [16b,16b+16); wave w
// covers cols [16w,16w+16). K-loop fully unrolled: 32 wmma ops per wave.
//
// ISA 7.12.2 layouts (wave32):
//   A 16x4 f32 : lane L<16 -> row L,    v0=K0,v1=K1 ; lane L>=16 -> row L-16, v0=K2,v1=K3
//   B 4x16 f32 : v0: lanes0-15 K=0 / lanes16-31 K=2 ; v1: K=1 / K=3 (col = lane&15)
//   C/D 16x16  : vgpr r: lanes0-15 -> M=r, N=lane ; lanes16-31 -> M=r+8, N=lane-16
__global__ __launch_bounds__(256) void k_gemm_wmma(const float* __restrict__ X,
                                                   const float* __restrict__ Wm,
                                                   float* __restrict__ H, int nrows) {
  const int lane   = threadIdx.x & 31;
  const int wave   = threadIdx.x >> 5;      // col tile 0..7
  const int laneLo = lane & 15;
  const int hi     = (lane >> 4) & 1;       // upper lane half -> K offset +2
  const int rowT   = blockIdx.x;

  int arow = rowT * 16 + laneLo;
  if (arow >= nrows) arow = nrows - 1;      // clamp (loads only; EXEC stays full for wmma)
  const float* aptr = X + (size_t)arow * NFEAT + 2 * hi;
  const int    bcol = wave * 16 + laneLo;

  v8f c = {};
#pragma unroll
  for (int k0 = 0; k0 < NFEAT; k0 += 4) {
    v2f a = *(const v2f*)(aptr + k0);               // A[row][k0+2*hi .. +1]
    v2f b;
    const int bk = k0 + 2 * hi;
    b.x = Wm[(size_t)bk * NFEAT + bcol];            // B[k0+2*hi][col]
    b.y = Wm[(size_t)(bk + 1) * NFEAT + bcol];      // B[k0+2*hi+1][col]
    c = __builtin_amdgcn_wmma_f32_16x16x4_f32(false, a, false, b,
                                              (short)0, c, false, false);
  }

  const int crowBase = rowT * 16 + 8 * hi;
  const int ccol     = wave * 16 + laneLo;
#pragma unroll
  for (int r = 0; r < 8; ++r) {
    int crow = crowBase + r;
    if (crow < nrows) H[(size_t)crow * NFEAT + ccol] = c[r];
  }
}

// ---- agg init: self-loop message + bias ------------------------------------
// agg[i][c] = b[c] + h[i][c] * dinv[i]^2
__global__ void k_init_agg(const float* __restrict__ h, const float* __restrict__ bias,
                           const float* __restrict__ dinv, float* __restrict__ agg, int n) {
  int idx = blockIdx.x * blockDim.x + threadIdx.x;
  if (idx < n * NFEAT) {
    int i = idx >> 7;            // NFEAT = 128
    int cc = idx & (NFEAT - 1);
    float di = dinv[i];
    agg[idx] = bias[cc] + h[idx] * di * di;
  }
}

// ---- edge scatter: agg[dst] += h[src] * norm -------------------------------
// One wave per edge; lane l handles features [4l, 4l+4). float4 gather from h,
// four f32 global atomics into agg. Both arrays are L2-resident (51.2MB each).
__global__ __launch_bounds__(256) void k_scatter(const int* __restrict__ src,
                                                 const int* __restrict__ dst,
                                                 const float* __restrict__ norm,
                                                 const float* __restrict__ h,
                                                 float* __restrict__ agg, int E) {
  const int e = blockIdx.x * 8 + (threadIdx.x >> 5);   // wave-uniform edge id
  if (e >= E) return;
  const int lane = threadIdx.x & 31;

  const int   s   = src[e];
  const int   d   = dst[e];
  const float nrm = norm[e];

  const float4 v = ((const float4*)(h + (size_t)s * NFEAT))[lane];
  float* a = agg + (size_t)d * NFEAT + lane * 4;
  atomicAdd(a + 0, v.x * nrm);
  atomicAdd(a + 1, v.y * nrm);
  atomicAdd(a + 2, v.z * nrm);
  atomicAdd(a + 3, v.w * nrm);
}

// ---- final relu ------------------------------------------------------------
__global__ void k_relu(const float* __restrict__ agg, float* __restrict__ out, int total) {
  int i = blockIdx.x * blockDim.x + threadIdx.x;
  if (i < total) out[i] = fmaxf(agg[i], 0.0f);
}

// ---------------------------------------------------------------------------
extern "C" void kernel_launch(void* const* d_in, const int* in_sizes, int n_in,
                              void* d_out, int out_size, void* d_ws, size_t ws_size,
                              hipStream_t stream) {
  const float* x  = (const float*)d_in[0];
  const int*   ei = (const int*)d_in[1];     // edge_index [2, E]
  const float* W1 = (const float*)d_in[2];
  const float* b1 = (const float*)d_in[3];
  const float* W2 = (const float*)d_in[4];
  const float* b2 = (const float*)d_in[5];
  float* out = (float*)d_out;

  const int N = in_sizes[0] / NFEAT;   // 100000
  const int E = in_sizes[1] / 2;       // 1600000
  const int* src = ei;
  const int* dst = ei + E;

  // workspace carve-up (all float4-aligned): dinv | norm | h | agg  (~109 MB)
  float* dinv = (float*)d_ws;                    // N     (deg, then dinv in place)
  float* norm = dinv + (size_t)N;                // E
  float* h    = norm + (size_t)E;                // N*128 (h1, then h2)
  float* agg  = h + (size_t)N * NFEAT;           // N*128 (agg1, then agg2)

  const int T = 256;
  const int gN  = (N + T - 1) / T;
  const int gE  = (E + T - 1) / T;
  const int gNF = (N * NFEAT + T - 1) / T;
  const int gRT = (N + 15) / 16;                 // gemm row tiles (6250)
  const int gSC = (E + 7) / 8;                   // scatter: 8 waves/block

  // ---- normalization (shared by both layers) ----
  k_init_deg<<<gN, T, 0, stream>>>(dinv, N);
  k_degree  <<<gE, T, 0, stream>>>(dst, dinv, E);
  k_dinv    <<<gN, T, 0, stream>>>(dinv, N);
  k_norm    <<<gE, T, 0, stream>>>(src, dst, dinv, norm, E);

  // ---- layer 1 ----
  k_gemm_wmma<<<gRT, T, 0, stream>>>(x, W1, h, N);
  k_init_agg <<<gNF, T, 0, stream>>>(h, b1, dinv, agg, N);
  k_scatter  <<<gSC, T, 0, stream>>>(src, dst, norm, h, agg, E);

  // ---- layer 2 (h buffer reused for h2, agg overwritten in-place) ----
  k_gemm_wmma<<<gRT, T, 0, stream>>>(agg, W2, h, N);
  k_init_agg <<<gNF, T, 0, stream>>>(h, b2, dinv, agg, N);
  k_scatter  <<<gSC, T, 0, stream>>>(src, dst, norm, h, agg, E);

  // ---- relu -> output ----
  k_relu<<<gNF, T, 0, stream>>>(agg, out, N * NFEAT);
}